// SplineModel_66159676228004
// MI455X (gfx1250) — compile-verified
//
#include <hip/hip_runtime.h>
#include <math.h>

typedef __attribute__((ext_vector_type(2))) float v2f;
typedef __attribute__((ext_vector_type(4))) float f4v;
typedef __attribute__((ext_vector_type(8))) float v8f;

namespace {
constexpr float kEps = 10.0f;
}

// D(16x16,f32) = A(16x4,f32) x B(4x16,f32) + C   -- CDNA5 V_WMMA_F32_16X16X4_F32
__device__ __forceinline__ v8f wmma4(v2f a, v2f b, v8f c) {
  return __builtin_amdgcn_wmma_f32_16x16x4_f32(
      /*neg_a=*/false, a, /*neg_b=*/false, b,
      /*c_mod=*/(short)0, c, /*reuse_a=*/false, /*reuse_b=*/false);
}

__device__ __forceinline__ float frcp(float x) { return __builtin_amdgcn_rcpf(x); }
__device__ __forceinline__ float fsqrt(float x) { return __builtin_amdgcn_sqrtf(x); }

// Build this lane's B-fragment of H = W (4x4) * Bv^T (4x16):
//   lanes 0-15  hold (H[0][j], H[1][j]) for column j = lane
//   lanes 16-31 hold (H[2][j], H[3][j]) for column j = lane-16
__device__ __forceinline__ v2f make_bfrag(const float* __restrict__ W16,
                                          const float4& bv, int k0) {
  const float* w0 = W16 + k0 * 4;
  v2f r;
  r.x = w0[0] * bv.x + w0[1] * bv.y + w0[2] * bv.z + w0[3] * bv.w;
  r.y = w0[4] * bv.x + w0[5] * bv.y + w0[6] * bv.z + w0[7] * bv.w;
  return r;
}

__global__ __launch_bounds__(128) void spline_wmma_kernel(
    const float* __restrict__ cp,      // [P,4,4,3]
    const float* __restrict__ wts,     // [P,4,4]
    const float* __restrict__ shdc,    // [P,4,4,1,3]
    const float* __restrict__ shrest,  // [P,4,4,15,3]
    const float* __restrict__ sp,      // [P,2]
    const float* __restrict__ opac,    // [N,1]
    const float* __restrict__ bu, const float* __restrict__ bvv,
    const float* __restrict__ dbu, const float* __restrict__ dbvv,
    const float* __restrict__ ddbu, const float* __restrict__ ddbvv,
    float* __restrict__ out, unsigned Nu) {
  __shared__ float s_w[16];        // nurbs weights [k*4+l]
  __shared__ float s_wcp[3][16];   // weighted control pts, per xyz component
  __shared__ float s_wcpsum[16];   // sum over xyz components (for curvature)
  __shared__ float s_shf[48][16];  // SH features, [s*3+c][k*4+l]
  __shared__ float s_red[4];       // per-wave area partials

  const int a = blockIdx.x;  // patch
  const int t = threadIdx.x; // 0..127
  const int lane = t & 31;
  const int wv = t >> 5;           // wave 0..3 -> 16x16 tile of the 32x32 grid
  const int ti = wv >> 1, tj = wv & 1;
  const int half = lane >> 4, lq = lane & 15;
  const int k0 = half * 2;

  // ---- cooperative per-patch staging into LDS ----
  const float* cpp = cp + (size_t)a * 48;
  const float* wp = wts + (size_t)a * 16;
  if (t < 16) s_w[t] = wp[t];
  if (t >= 16 && t < 64) {
    int e = t - 16, c = e >> 4, kl = e & 15;
    s_wcp[c][kl] = cpp[kl * 3 + c] * wp[kl];
  }
  for (int e = t; e < 768; e += 128) {  // 48 components x 16 ctrl pts
    int sc = e >> 4, kl = e & 15;
    int s = sc / 3, c = sc - 3 * s;
    float v = (s == 0) ? shdc[(size_t)a * 48 + kl * 3 + c]
                       : shrest[(size_t)a * 720 + kl * 45 + (s - 1) * 3 + c];
    s_shf[sc][kl] = v;
  }
  __syncthreads();
  if (t < 16) s_wcpsum[t] = s_wcp[0][t] + s_wcp[1][t] + s_wcp[2][t];
  __syncthreads();

  // ---- per-lane WMMA fragments (A from *_u rows, Bv rows from *_v) ----
  const size_t bb = (size_t)a * 128;
  const int arow = 16 * ti + lq;   // A: M across lanes 0-15, K pair by half
  v2f Ab   = *(const v2f*)(bu   + bb + arow * 4 + k0);
  v2f Adb  = *(const v2f*)(dbu  + bb + arow * 4 + k0);
  v2f Addb = *(const v2f*)(ddbu + bb + arow * 4 + k0);
  const int jcol = 16 * tj + lq;   // B: N across lanes 0-15, K pair by half
  float4 bv4   = *(const float4*)(bvv   + bb + jcol * 4);
  float4 dbv4  = *(const float4*)(dbvv  + bb + jcol * 4);
  float4 ddbv4 = *(const float4*)(ddbvv + bb + jcol * 4);

  v8f zero = {};

  // ---- NURBS denominator (guard den==0 -> 1), keep reciprocal ----
  v8f den = wmma4(Ab, make_bfrag(s_w, bv4, k0), zero);
  float dinv[8];
#pragma unroll
  for (int r = 0; r < 8; ++r) {
    float d = den[r];
    d = (d == 0.0f) ? 1.0f : d;
    dinv[r] = frcp(d);
  }

  // flat point indices (32-bit: out has 59*N < 2^31 elements)
  unsigned nidx[8];
#pragma unroll
  for (int r = 0; r < 8; ++r) {
    int i = 16 * ti + r + 8 * half;     // C/D layout: M = r (+8 for hi half)
    nidx[r] = (unsigned)a * 1024u + (unsigned)i * 32u + (unsigned)jcol;
  }

  float* const outXyz = out;
  float* const outOp  = out + (size_t)3 * Nu;
  float* const outScl = out + (size_t)4 * Nu;
  float* const outRot = out + (size_t)7 * Nu;
  float* const outShs = out + (size_t)11 * Nu;

  // ---- xyz: compute, divide, stream out (NT: write-once data) ----
#pragma unroll
  for (int c = 0; c < 3; ++c) {
    v8f x = wmma4(Ab, make_bfrag(s_wcp[c], bv4, k0), zero);
#pragma unroll
    for (int r = 0; r < 8; ++r)
      __builtin_nontemporal_store(x[r] * dinv[r], outXyz + nidx[r] * 3u + c);
  }

  // ---- first derivatives (kept for cross/scaling) ----
  v8f du0 = wmma4(Adb, make_bfrag(s_wcp[0], bv4, k0), zero);
  v8f du1 = wmma4(Adb, make_bfrag(s_wcp[1], bv4, k0), zero);
  v8f du2 = wmma4(Adb, make_bfrag(s_wcp[2], bv4, k0), zero);
  v8f dv0 = wmma4(Ab, make_bfrag(s_wcp[0], dbv4, k0), zero);
  v8f dv1 = wmma4(Ab, make_bfrag(s_wcp[1], dbv4, k0), zero);
  v8f dv2 = wmma4(Ab, make_bfrag(s_wcp[2], dbv4, k0), zero);
  // second-derivative component SUMS only (curvature needs ddu.sum(-1)):
  v8f dduS = wmma4(Addb, make_bfrag(s_wcpsum, bv4, k0), zero);
  v8f ddvS = wmma4(Ab, make_bfrag(s_wcpsum, ddbv4, k0), zero);

  // ---- per-point vector math: cross, rotation, opacity, area partial ----
  float sxv[8], syv[8];
  float csum = 0.0f;
#pragma unroll
  for (int r = 0; r < 8; ++r) {
    float dux = du0[r] * dinv[r], duy = du1[r] * dinv[r], duz = du2[r] * dinv[r];
    float dvx = dv0[r] * dinv[r], dvy = dv1[r] * dinv[r], dvz = dv2[r] * dinv[r];
    float cx = duy * dvz - duz * dvy;
    float cy = duz * dvx - dux * dvz;
    float cz = dux * dvy - duy * dvx;
    csum += fsqrt(cx * cx + cy * cy + cz * cz);
    // L1 normalize then L2 normalize (matches reference chain)
    float r1 = frcp(fmaxf(fabsf(cx) + fabsf(cy) + fabsf(cz), 1e-12f));
    float n1x = cx * r1, n1y = cy * r1, n1z = cz * r1;
    float r2 = frcp(fmaxf(fsqrt(n1x * n1x + n1y * n1y + n1z * n1z), 1e-12f));
    float nx = n1x * r2, ny = n1y * r2, nz = n1z * r2;
    float qw = 1.0f + nz;
    float rq = frcp(fmaxf(fsqrt(qw * qw + nx * nx + ny * ny), 1e-12f));
    unsigned n = nidx[r];
    f4v q = {qw * rq, -ny * rq, nx * rq, 0.0f};
    __builtin_nontemporal_store(q, (f4v*)(outRot + n * 4u));
    __builtin_nontemporal_store(__builtin_nontemporal_load(opac + n), outOp + n);
    sxv[r] = dux * dux + duy * duy + duz * duz + kEps;
    syv[r] = dvx * dvx + dvy * dvy + dvz * dvz + kEps;
  }

  // ---- per-patch area mean: wave shuffle reduce + LDS combine ----
  float v = csum;
#pragma unroll
  for (int o = 16; o > 0; o >>= 1) v += __shfl_xor(v, o, 32);
  if (lane == 0) s_red[wv] = v;
  __syncthreads();
  float area_mean = (s_red[0] + s_red[1] + s_red[2] + s_red[3]) * (1.0f / 1024.0f);

  // ---- scaling ----
  float sfb = area_mean * (1.0f / 1024.0f);  // area / R^2
  float sf0 = sfb * sp[(size_t)a * 2 + 0];
  float sf1 = sfb * sp[(size_t)a * 2 + 1];
  float sf2 = sfb * 1e-16f;
  float sc2 = 0.5f * __logf(fabsf(sf2 * kEps)) - __logf(1.0f + kEps);
#pragma unroll
  for (int r = 0; r < 8; ++r) {
    float cu = dduS[r] * dinv[r], cv = ddvS[r] * dinv[r];
    unsigned n = nidx[r];
    float* s3 = outScl + n * 3u;
    __builtin_nontemporal_store(
        0.5f * __logf(fabsf(sf0 * sxv[r])) - __logf(fabsf(cu) + kEps), s3 + 0);
    __builtin_nontemporal_store(
        0.5f * __logf(fabsf(sf1 * syv[r])) - __logf(fabsf(cv) + kEps), s3 + 1);
    __builtin_nontemporal_store(sc2, s3 + 2);
  }

  // ---- SH synthesis: 48 comps, 4 at a time -> b128 NT stores (no den) ----
  for (int g = 0; g < 12; ++g) {
    v8f d0 = wmma4(Ab, make_bfrag(s_shf[4 * g + 0], bv4, k0), zero);
    v8f d1 = wmma4(Ab, make_bfrag(s_shf[4 * g + 1], bv4, k0), zero);
    v8f d2 = wmma4(Ab, make_bfrag(s_shf[4 * g + 2], bv4, k0), zero);
    v8f d3 = wmma4(Ab, make_bfrag(s_shf[4 * g + 3], bv4, k0), zero);
#pragma unroll
    for (int r = 0; r < 8; ++r) {
      f4v q = {d0[r], d1[r], d2[r], d3[r]};  // 16B-aligned: n*192 + g*16
      __builtin_nontemporal_store(q, (f4v*)(outShs + nidx[r] * 48u + 4u * g));
    }
  }
}

extern "C" void kernel_launch(void* const* d_in, const int* in_sizes, int n_in,
                              void* d_out, int out_size, void* d_ws, size_t ws_size,
                              hipStream_t stream) {
  const float* cp     = (const float*)d_in[0];
  const float* wts    = (const float*)d_in[1];
  const float* shdc   = (const float*)d_in[2];
  const float* shrest = (const float*)d_in[3];
  const float* sp     = (const float*)d_in[4];
  const float* opac   = (const float*)d_in[5];
  const float* bu     = (const float*)d_in[6];
  const float* bv     = (const float*)d_in[7];
  const float* dbu    = (const float*)d_in[8];
  const float* dbv    = (const float*)d_in[9];
  const float* ddbu   = (const float*)d_in[10];
  const float* ddbv   = (const float*)d_in[11];

  const int P = in_sizes[1] / 16;                 // nurbs_weights = P*4*4
  const unsigned N = (unsigned)P * 1024u;

  spline_wmma_kernel<<<P, 128, 0, stream>>>(
      cp, wts, shdc, shrest, sp, opac, bu, bv, dbu, dbv, ddbu, ddbv,
      (float*)d_out, N);
}